// Model_79147657330978
// MI455X (gfx1250) — compile-verified
//
#include <hip/hip_runtime.h>

#define N_NODES 50000
#define N_HE    10000
#define N_EDGE  800000
#define N_INC   800000
#define HID     128
#define OUTC    64
#define SLOPE   0.01f

typedef __attribute__((ext_vector_type(2))) float v2f;
typedef __attribute__((ext_vector_type(8))) float v8f;

// ---------------------------------------------------------------------------
// GEMM: Y[M x (NT*16)] = A[M x 128] * W[128 x (NT*16)]  (+add +bias, leaky)
// fp32 WMMA 16x16x4. 4 waves/block, 16 rows/wave, NT column tiles/wave.
// W is staged into LDS *pre-swizzled into WMMA B-fragment layout* so the
// inner loop issues a single ds_load_b64 per fragment (conflict-free).
// Epilogue behavior is compile-time templated; bounds check is one uniform
// branch (only the last partial block takes the guarded path).
// ---------------------------------------------------------------------------
template <int NT, bool HAS_ADD, bool HAS_BIAS, bool LEAKY>
__launch_bounds__(128)
__global__ void gemm_wmma_f32(const float* __restrict__ A,
                              const float* __restrict__ W,
                              float* __restrict__ Y,
                              const float* __restrict__ addbuf, // M x NCOLS
                              const float* __restrict__ bias,   // NCOLS
                              int M) {
  constexpr int NCOLS  = NT * 16;
  constexpr int K      = 128;
  constexpr int KSTEPS = K / 4;                 // 32
  __shared__ v2f sW[KSTEPS * NT * 32];          // fragment-layout W (64KB @ NT=8)

  // Stage W swizzled: sW[(ks*NT + t)*32 + lane] = {W[k][col], W[k+1][col]}
  // with k = 4*ks + 2*(lane>=16), col = t*16 + (lane&15).
  for (int i = threadIdx.x; i < KSTEPS * NT * 32; i += blockDim.x) {
    const int ln   = i & 31;
    const int tile = (i >> 5) % NT;
    const int ks   = (i >> 5) / NT;
    const int k    = ks * 4 + ((ln >> 4) << 1);
    const int col  = tile * 16 + (ln & 15);
    v2f w;
    w.x = W[k * NCOLS + col];
    w.y = W[(k + 1) * NCOLS + col];
    sW[i] = w;
  }
  __syncthreads();

  const int wave = threadIdx.x >> 5;
  const int lane = threadIdx.x & 31;
  const int row0 = (blockIdx.x * 4 + wave) * 16;
  const int m16  = lane & 15;
  const int koff = (lane >> 4) << 1;

  // clamp A row so loads stay in-bounds with EXEC all-ones (WMMA requirement)
  int arow = row0 + m16;
  arow = arow < M ? arow : (M - 1);
  const float* aptr = A + (size_t)arow * K + koff;

  v8f c[NT];
  const v8f zero8 = {0.f, 0.f, 0.f, 0.f, 0.f, 0.f, 0.f, 0.f};
#pragma unroll
  for (int t = 0; t < NT; ++t) c[t] = zero8;

#pragma unroll 4
  for (int ks = 0; ks < KSTEPS; ++ks) {
    const v2f a = *(const v2f*)(aptr + ks * 4);      // single global_load_b64
#pragma unroll
    for (int t = 0; t < NT; ++t) {
      const v2f b = sW[(ks * NT + t) * 32 + lane];   // single ds_load_b64
      c[t] = __builtin_amdgcn_wmma_f32_16x16x4_f32(
          false, a, false, b, (short)0, c[t], false, false);
    }
  }

  // epilogue: D layout -> VGPR r holds row (r + 8*(lane>=16)), col = lane&15
  const int rbase = row0 + ((lane >> 4) << 3);
  float bcol[NT];
  if (HAS_BIAS) {
#pragma unroll
    for (int t = 0; t < NT; ++t) bcol[t] = bias[t * 16 + m16];
  }

  if (row0 + 16 <= M) {                 // uniform: full tile, no guards
#pragma unroll
    for (int t = 0; t < NT; ++t) {
      const int col = t * 16 + m16;
#pragma unroll
      for (int r = 0; r < 8; ++r) {
        const size_t off = (size_t)(rbase + r) * NCOLS + col;
        float v = c[t][r];
        if (HAS_ADD)  v += addbuf[off];
        if (HAS_BIAS) v += bcol[t];
        if (LEAKY)    v = (v >= 0.f) ? v : SLOPE * v;
        Y[off] = v;
      }
    }
  } else {                              // last partial block only
#pragma unroll
    for (int t = 0; t < NT; ++t) {
      const int col = t * 16 + m16;
#pragma unroll
      for (int r = 0; r < 8; ++r) {
        const int row = rbase + r;
        if (row < M) {
          const size_t off = (size_t)row * NCOLS + col;
          float v = c[t][r];
          if (HAS_ADD)  v += addbuf[off];
          if (HAS_BIAS) v += bcol[t];
          if (LEAKY)    v = (v >= 0.f) ? v : SLOPE * v;
          Y[off] = v;
        }
      }
    }
  }
}

// ---------------------------------------------------------------------------
// One incidence per wave32; each lane moves float4 (128 floats per row).
// dst[scatter_idx[i]] += src[gather_idx[i]]   (rows of width 128)
// ---------------------------------------------------------------------------
__global__ void scatter_add_rows(const float* __restrict__ src,
                                 float* __restrict__ dst,
                                 const int* __restrict__ gather_idx,
                                 const int* __restrict__ scatter_idx, int n) {
  const int inc  = blockIdx.x * (blockDim.x >> 5) + (threadIdx.x >> 5);
  const int lane = threadIdx.x & 31;
  if (inc >= n) return;
  const int s = gather_idx[inc];
  const int d = scatter_idx[inc];
  const float4 v = ((const float4*)(src + (size_t)s * HID))[lane];
  float* dp = dst + (size_t)d * HID + lane * 4;
  atomicAdd(dp + 0, v.x);
  atomicAdd(dp + 1, v.y);
  atomicAdd(dp + 2, v.z);
  atomicAdd(dp + 3, v.w);
}

__global__ void degree_kernel(const int* __restrict__ node_idx,
                              const int* __restrict__ he_idx,
                              float* __restrict__ D, float* __restrict__ B, int n) {
  const int i = blockIdx.x * blockDim.x + threadIdx.x;
  if (i < n) {
    atomicAdd(&D[node_idx[i]], 1.0f);
    atomicAdd(&B[he_idx[i]], 1.0f);
  }
}

// m[row][col] *= (B[row] > 0 ? 1/B[row] : 0)
__global__ void scale_rows_inv(float* __restrict__ m, const float* __restrict__ B,
                               int total) {
  const int i = blockIdx.x * blockDim.x + threadIdx.x;
  if (i < total) {
    const float b = B[i >> 7];
    m[i] = (b > 0.f) ? m[i] / b : 0.f;
  }
}

// Xnew = leaky( o * Dinv + l + hb + lb )
__global__ void hconv_finalize(float* __restrict__ Xnew, const float* __restrict__ o,
                               const float* __restrict__ D, const float* __restrict__ l,
                               const float* __restrict__ hb, const float* __restrict__ lb,
                               int total) {
  const int i = blockIdx.x * blockDim.x + threadIdx.x;
  if (i >= total) return;
  const int row = i >> 7, col = i & 127;
  const float d = D[row];
  float v = (d > 0.f ? o[i] / d : 0.f) + l[i] + hb[col] + lb[col];
  Xnew[i] = (v >= 0.f) ? v : SLOPE * v;
}

__global__ void add_vec(float* __restrict__ out, const float* __restrict__ a,
                        const float* __restrict__ b, int n) {
  const int i = blockIdx.x * blockDim.x + threadIdx.x;
  if (i < n) out[i] = a[i] + b[i];
}

// ---------------------------------------------------------------------------
// Host-side layer drivers (all launches on `stream`; graph-capture safe)
// ---------------------------------------------------------------------------
static void run_hconv(const float* Xin, const float* hw, const float* hb,
                      const float* lw, const float* lb, float* Xout,
                      float* t, float* l, float* m, const float* Dd, const float* Bd,
                      const int* node_idx, const int* he_idx, hipStream_t stream) {
  const int ggrd = (N_NODES + 63) / 64;
  gemm_wmma_f32<8, false, false, false><<<ggrd, 128, 0, stream>>>(
      Xin, hw, t, nullptr, nullptr, N_NODES);
  gemm_wmma_f32<8, false, false, false><<<ggrd, 128, 0, stream>>>(
      Xin, lw, l, nullptr, nullptr, N_NODES);
  hipMemsetAsync(m, 0, (size_t)N_HE * HID * sizeof(float), stream);
  scatter_add_rows<<<(N_INC + 7) / 8, 256, 0, stream>>>(t, m, node_idx, he_idx, N_INC);
  scale_rows_inv<<<(N_HE * HID + 255) / 256, 256, 0, stream>>>(m, Bd, N_HE * HID);
  hipMemsetAsync(t, 0, (size_t)N_NODES * HID * sizeof(float), stream);
  scatter_add_rows<<<(N_INC + 7) / 8, 256, 0, stream>>>(m, t, he_idx, node_idx, N_INC);
  hconv_finalize<<<(N_NODES * HID + 255) / 256, 256, 0, stream>>>(
      Xout, t, Dd, l, hb, lb, N_NODES * HID);
}

static void run_gconv(const float* Xin, const float* wrel, const float* brel,
                      const float* wroot, const float* gw, const float* gb,
                      float* Xout, float* t, float* l, float* wcomb, float* bcomb,
                      const int* src_idx, const int* dst_idx, hipStream_t stream) {
  hipMemsetAsync(t, 0, (size_t)N_NODES * HID * sizeof(float), stream);
  scatter_add_rows<<<(N_EDGE + 7) / 8, 256, 0, stream>>>(Xin, t, src_idx, dst_idx, N_EDGE);
  const int ggrd = (N_NODES + 63) / 64;
  gemm_wmma_f32<8, false, false, false><<<ggrd, 128, 0, stream>>>(
      t, wrel, l, nullptr, nullptr, N_NODES);
  add_vec<<<(HID * HID + 255) / 256, 256, 0, stream>>>(wcomb, wroot, gw, HID * HID);
  add_vec<<<1, 128, 0, stream>>>(bcomb, brel, gb, HID);
  gemm_wmma_f32<8, true, true, true><<<ggrd, 128, 0, stream>>>(
      Xin, wcomb, Xout, l, bcomb, N_NODES);
}

extern "C" void kernel_launch(void* const* d_in, const int* in_sizes, int n_in,
                              void* d_out, int out_size, void* d_ws, size_t ws_size,
                              hipStream_t stream) {
  const float* X0      = (const float*)d_in[0];
  const int*   edge    = (const int*)d_in[1];   // [2 x E]: row0 src, row1 dst
  const int*   hinc    = (const int*)d_in[2];   // [2 x INC]: row0 node, row1 he
  const float* hc1_w   = (const float*)d_in[3];
  const float* hc1_b   = (const float*)d_in[4];
  const float* lin1_w  = (const float*)d_in[5];
  const float* lin1_b  = (const float*)d_in[6];
  const float* hc2_w   = (const float*)d_in[7];
  const float* hc2_b   = (const float*)d_in[8];
  const float* lin2_w  = (const float*)d_in[9];
  const float* lin2_b  = (const float*)d_in[10];
  const float* c1_wrel = (const float*)d_in[11];
  const float* c1_brel = (const float*)d_in[12];
  const float* c1_wroot= (const float*)d_in[13];
  const float* glin1_w = (const float*)d_in[14];
  const float* glin1_b = (const float*)d_in[15];
  const float* c2_wrel = (const float*)d_in[16];
  const float* c2_brel = (const float*)d_in[17];
  const float* c2_wroot= (const float*)d_in[18];
  const float* glin2_w = (const float*)d_in[19];
  const float* glin2_b = (const float*)d_in[20];
  const float* fc_w    = (const float*)d_in[21];
  const float* fc_b    = (const float*)d_in[22];

  // workspace layout (floats): ~82 MB total
  float* ws    = (float*)d_ws;
  float* t     = ws;                                  // N*128 (temp / agg / scatter dst)
  float* l     = t + (size_t)N_NODES * HID;           // N*128 (lin / wrel output)
  float* Xb    = l + (size_t)N_NODES * HID;           // N*128 (X ping)
  float* m     = Xb + (size_t)N_NODES * HID;          // NUM_HE*128
  float* Dd    = m + (size_t)N_HE * HID;              // N
  float* Bd    = Dd + N_NODES;                        // NUM_HE
  float* wcomb = Bd + N_HE;                           // 128*128
  float* bcomb = wcomb + HID * HID;                   // 128

  float* Xq   = (float*)d_out;                        // X pong -> final X region
  float* Yout = Xq + (size_t)N_NODES * HID;           // y region

  const int* node_idx = hinc;
  const int* he_idx   = hinc + N_INC;
  const int* src_idx  = edge;
  const int* dst_idx  = edge + N_EDGE;

  // node / hyperedge degrees (shared by both hconv layers)
  hipMemsetAsync(Dd, 0, N_NODES * sizeof(float), stream);
  hipMemsetAsync(Bd, 0, N_HE * sizeof(float), stream);
  degree_kernel<<<(N_INC + 255) / 256, 256, 0, stream>>>(node_idx, he_idx, Dd, Bd, N_INC);

  // layer 1: hconv + lin  (X0 -> Xb)
  run_hconv(X0, hc1_w, hc1_b, lin1_w, lin1_b, Xb, t, l, m, Dd, Bd,
            node_idx, he_idx, stream);
  // layer 2: hconv + lin  (Xb -> Xq)
  run_hconv(Xb, hc2_w, hc2_b, lin2_w, lin2_b, Xq, t, l, m, Dd, Bd,
            node_idx, he_idx, stream);
  // layer 3: gconv + lin  (Xq -> Xb)
  run_gconv(Xq, c1_wrel, c1_brel, c1_wroot, glin1_w, glin1_b, Xb,
            t, l, wcomb, bcomb, src_idx, dst_idx, stream);
  // layer 4: gconv + lin  (Xb -> Xq == d_out X region)
  run_gconv(Xb, c2_wrel, c2_brel, c2_wroot, glin2_w, glin2_b, Xq,
            t, l, wcomb, bcomb, src_idx, dst_idx, stream);

  // final: y = X @ fc_w + fc_b  (N x 64)
  gemm_wmma_f32<4, false, true, false><<<(N_NODES + 63) / 64, 128, 0, stream>>>(
      Xq, fc_w, Yout, nullptr, fc_b, N_NODES);
}